// BilateralDenoiser_44427141710593
// MI455X (gfx1250) — compile-verified
//
#include <hip/hip_runtime.h>
#include <math.h>

#define IMG_H 1080
#define IMG_W 1920
#define NCH   11
#define TS    16
#define RAD   7
#define KDIM  (2*RAD + 1)          // 15
#define NTAP  (KDIM*KDIM)          // 225
#define HALO  (TS + 2*RAD)         // 30
#define EPSV  0.0001f
#define RCP_EPS 10000.0f           // 1/EPSV

typedef unsigned int u32x4 __attribute__((ext_vector_type(4)));
typedef int          i32x4 __attribute__((ext_vector_type(4)));
typedef int          i32x8 __attribute__((ext_vector_type(8)));

#if defined(__has_builtin)
#if __has_builtin(__builtin_amdgcn_tensor_load_to_lds) && __has_builtin(__builtin_amdgcn_s_wait_tensorcnt)
#define USE_TDM 1
#endif
#endif
#ifndef USE_TDM
#define USE_TDM 0
#endif

// Per-pixel bilateral accumulation over the 15x15 window, reading the halo
// tile from LDS. INTERIOR=true: all taps valid, tile dims are compile-time
// constants (LDS addresses fold to immediate ds offsets).
template<bool INTERIOR>
__device__ __forceinline__ void bilateral_compute(
    const float* __restrict__ tile,
    const float* __restrict__ wxyT,
    const float* __restrict__ rdistT,
    const int oy, const int ox,
    const int ry0, const int rx0,
    const int ldsW_in, const int ldsH_in,
    float* __restrict__ out)
{
    const int ldsW = INTERIOR ? HALO : ldsW_in;
    const int ldsH = INTERIOR ? HALO : ldsH_in;
    const int rowF = ldsW * NCH;
    const int cly = oy - ry0;
    const int clx = ox - rx0;

    const float* cp = &tile[(cly * ldsW + clx) * NCH];
    const float n0 = cp[3], n1 = cp[4], n2 = cp[5];
    const float zc = cp[9], dzc = cp[10];
    // 1/max(dzc*dist, eps) == min(rdzc * (1/dist), 1/eps)  (see analysis)
    const float rdzc = (dzc > 0.0f) ? __fdividef(1.0f, dzc) : __builtin_inff();

    float a0 = 0.f, a1 = 0.f, a2 = 0.f, aw = 0.f;

    #pragma unroll 1
    for (int dy = -RAD; dy <= RAD; ++dy) {
        int iy = cly + dy;
        bool vy = true;
        if (!INTERIOR) {
            vy = ((unsigned)(oy + dy) < (unsigned)IMG_H);
            iy = iy < 0 ? 0 : (iy >= ldsH ? ldsH - 1 : iy);
        }
        const float* rowp = &tile[iy * rowF + clx * NCH];   // tap dx=0
        const float* wrow = &wxyT[(dy + RAD) * KDIM + RAD];
        const float* rrow = &rdistT[(dy + RAD) * KDIM + RAD];

        #pragma unroll
        for (int dx = -RAD; dx <= RAD; ++dx) {
            const float* tp;
            bool v = vy;
            if (INTERIOR) {
                tp = rowp + dx * NCH;
            } else {
                v = vy & ((unsigned)(ox + dx) < (unsigned)IMG_W);
                int ix = clx + dx;
                ix = ix < 0 ? 0 : (ix >= ldsW ? ldsW - 1 : ix);
                tp = &tile[iy * rowF + ix * NCH];
            }

            const float c0 = tp[0], c1 = tp[1], c2 = tp[2];
            const float dot = fmaf(tp[3], n0, fmaf(tp[4], n1, tp[5] * n2));
            const float zt  = tp[9];

            // w_normal = clip(dot, eps, 1)^128 -> 7 squarings
            float d = dot < EPSV ? EPSV : (dot > 1.0f ? 1.0f : dot);
            d = d * d; d = d * d; d = d * d; d = d * d;
            d = d * d; d = d * d; d = d * d;

            // w_depth = exp(-|zt-zc| * min(rdzc/dist, 1/eps))
            float s = rdzc * rrow[dx];
            s = (s < RCP_EPS) ? s : RCP_EPS;     // inf (dz<=0 or dist==0) -> 1/eps
            const float wd = __expf(-fabsf(zt - zc) * s);

            float w = wrow[dx] * d * wd;
            if (!INTERIOR) w = v ? w : 0.0f;

            a0 = fmaf(c0, w, a0);
            a1 = fmaf(c1, w, a1);
            a2 = fmaf(c2, w, a2);
            aw += w;
        }
    }

    const float inv = __fdividef(1.0f, aw < EPSV ? EPSV : aw);
    float* op = out + ((size_t)oy * IMG_W + ox) * 3;
    op[0] = a0 * inv;
    op[1] = a1 * inv;
    op[2] = a2 * inv;
}

__launch_bounds__(256)
__global__ void BilateralDenoiser_kernel(const float* __restrict__ in,
                                         float* __restrict__ out) {
    __shared__ float tile[HALO * HALO * NCH];   // 39600 bytes, TDM-packed rows
    __shared__ float wxyT[NTAP];
    __shared__ float rdistT[NTAP];

    const int tid = threadIdx.x;
    const int ox0 = blockIdx.x * TS;
    const int oy0 = blockIdx.y * TS;

    int rx0 = ox0 - RAD; if (rx0 < 0) rx0 = 0;
    int ry0 = oy0 - RAD; if (ry0 < 0) ry0 = 0;
    int rx1 = ox0 + TS + RAD; if (rx1 > IMG_W) rx1 = IMG_W;
    int ry1 = oy0 + TS + RAD; if (ry1 > IMG_H) ry1 = IMG_H;
    const int ldsW = rx1 - rx0;
    const int ldsH = ry1 - ry0;
    const int rowFloats = ldsW * NCH;           // <= 330

#if USE_TDM
    if ((tid >> 5) == 0) {                      // one wave issues the DMA
        const unsigned lds_base =
            (unsigned)(unsigned long long)&tile[0];        // low 32 bits = LDS byte offset
        const unsigned long long gaddr =
            (unsigned long long)(in + ((size_t)ry0 * IMG_W + rx0) * NCH);

        // D# group 0: count=1, lds_addr, 57-bit global addr, type=2
        u32x4 g0;
        g0[0] = 1u;
        g0[1] = lds_base;
        g0[2] = (unsigned)(gaddr & 0xFFFFFFFFull);
        g0[3] = (unsigned)((gaddr >> 32) & 0x01FFFFFFull) | 0x80000000u;

        // D# group 1: 2D fp32 tensor, rows of rowFloats elems, ldsH rows,
        // row stride IMG_W*NCH elements.
        const unsigned td0   = (unsigned)rowFloats;
        const unsigned td1   = (unsigned)ldsH;
        const unsigned tile0 = (unsigned)rowFloats;
        const unsigned tile1 = (unsigned)ldsH;
        const unsigned long long s0 = (unsigned long long)IMG_W * NCH;

        i32x8 g1;
        g1[0] = (int)(2u << 16);                                    // data_size=4B
        g1[1] = (int)((td0 & 0xFFFFu) << 16);                       // abar=0 | dim0[15:0]
        g1[2] = (int)((td0 >> 16) | ((td1 & 0xFFFFu) << 16));       // dim0[31:16] | dim1[15:0]
        g1[3] = (int)((td1 >> 16) | ((tile0 & 0xFFFFu) << 16));     // dim1[31:16] | tile_dim0
        g1[4] = (int)(tile1 & 0xFFFFu);                             // tile_dim1 | tile_dim2=0
        g1[5] = (int)(s0 & 0xFFFFFFFFull);                          // dim0_stride[31:0]
        g1[6] = (int)((s0 >> 32) & 0xFFFFull);                      // stride[47:32] | d1s=0
        g1[7] = 0;

        i32x4 gz = {0, 0, 0, 0};
#if defined(__clang_major__) && (__clang_major__ >= 23)
        i32x8 gz8 = {0, 0, 0, 0, 0, 0, 0, 0};
        __builtin_amdgcn_tensor_load_to_lds(g0, g1, gz, gz, gz8, 0);
#else
        __builtin_amdgcn_tensor_load_to_lds(g0, g1, gz, gz, 0);
#endif
    }
#endif

    // Spatial Gaussian + reciprocal-distance tables while the DMA runs.
    if (tid < NTAP) {
        const int dy = tid / KDIM - RAD;
        const int dx = tid % KDIM - RAD;
        const float d2 = (float)(dy * dy + dx * dx);
        wxyT[tid]   = __expf(-0.5f * d2);        // VARIANCE = 1
        rdistT[tid] = __frsqrt_rn(d2);           // 1/dist; d2==0 -> +inf
    }

#if USE_TDM
    if ((tid >> 5) == 0) {
        __builtin_amdgcn_s_wait_tensorcnt(0);
    }
#else
    {   // Fallback: cooperative load of the packed halo tile.
        const int total = ldsH * rowFloats;
        const float* src = in + ((size_t)ry0 * IMG_W + rx0) * NCH;
        for (int q = tid; q < total; q += 256) {
            const int iy = q / rowFloats;
            const int r  = q - iy * rowFloats;
            tile[q] = src[(size_t)iy * (IMG_W * NCH) + r];
        }
    }
#endif
    __syncthreads();

    const int oy = oy0 + (tid >> 4);
    const int ox = ox0 + (tid & 15);
    if (oy >= IMG_H) return;                     // IMG_W % TS == 0, ox always valid

    const bool interior = (ox0 >= RAD) && (oy0 >= RAD) &&
                          (ox0 + TS + RAD <= IMG_W) && (oy0 + TS + RAD <= IMG_H);
    if (interior)
        bilateral_compute<true >(tile, wxyT, rdistT, oy, ox, ry0, rx0, ldsW, ldsH, out);
    else
        bilateral_compute<false>(tile, wxyT, rdistT, oy, ox, ry0, rx0, ldsW, ldsH, out);
}

extern "C" void kernel_launch(void* const* d_in, const int* in_sizes, int n_in,
                              void* d_out, int out_size, void* d_ws, size_t ws_size,
                              hipStream_t stream) {
    (void)in_sizes; (void)n_in; (void)d_ws; (void)ws_size; (void)out_size;
    const float* in = (const float*)d_in[0];
    float* out = (float*)d_out;
    dim3 grid((IMG_W + TS - 1) / TS, (IMG_H + TS - 1) / TS);  // 120 x 68
    BilateralDenoiser_kernel<<<grid, 256, 0, stream>>>(in, out);
}